// MSDeformableAttention_14860586844346
// MI455X (gfx1250) — compile-verified
//
#include <hip/hip_runtime.h>
#include <cmath>

typedef float v2f __attribute__((ext_vector_type(2)));
typedef float v8f __attribute__((ext_vector_type(8)));

namespace {
constexpr int kEmbed  = 256;
constexpr int kNH     = 8;
constexpr int kHD     = 32;
constexpr int kLevels = 4;
constexpr int kP      = 4;
constexpr int kSumP   = 16;
constexpr int kBS     = 4;
constexpr int kLQ     = 4096;
constexpr int kLV     = 8500;   // 80*80 + 40*40 + 20*20 + 10*10
}

static __device__ __forceinline__ int iclamp(int v, int lo, int hi) {
  return v < lo ? lo : (v > hi ? hi : v);
}

// ---------------------------------------------------------------------------
// Y[M,N] = A[M,256] @ W[256,N] + bias[N], fp32, via V_WMMA_F32_16X16X4_F32.
// Block = 256 threads = 8 waves. blockIdx.x selects the 16-col N tile (B tile
// staged once into LDS, 8x reuse); blockIdx.y*8 + wave selects the 16-row M
// tile. Each wave runs 64 WMMAs over K=256.
//
// ISA layouts (CDNA5 ISA, 32-bit):
//   A 16x4 : lanes 0-15 hold rows M=0..15 with K={k,k+1}; lanes 16-31 same
//            rows with K={k+2,k+3}  -> one b64 load per lane per k-step.
//   B 4x16 : mirrored (lane = N, half-wave = K pair).
//   C 16x16: VGPR r -> M=r (lanes 0-15) / M=r+8 (lanes 16-31), N=lane&15.
// ---------------------------------------------------------------------------
__global__ __launch_bounds__(256) void gemm_k256_wmma(
    const float* __restrict__ A, const float* __restrict__ W,
    const float* __restrict__ bias, float* __restrict__ Y,
    int M, int N)
{
  __shared__ float ldsB[kEmbed * 16];   // 16 KB: B tile, K-major

  const int n0 = blockIdx.x * 16;
  for (int t = threadIdx.x; t < kEmbed * 16; t += 256) {
    const int k = t >> 4;
    const int c = t & 15;
    ldsB[t] = W[k * N + n0 + c];
  }
  __syncthreads();

  const int tile_m = blockIdx.y * 8 + (threadIdx.x >> 5);   // wave-uniform
  if (tile_m * 16 >= M) return;                              // after barrier: safe

  const int lane = threadIdx.x & 31;
  const int hi = lane >> 4;       // half-wave selects K pair
  const int ln = lane & 15;       // row (A) / col (B,C)

  const float* __restrict__ arow = A + (size_t)(tile_m * 16 + ln) * kEmbed + 2 * hi;
  const float* __restrict__ bcol = ldsB + (2 * hi) * 16 + ln;

  v8f acc = {};
#pragma unroll 8
  for (int k = 0; k < kEmbed; k += 4) {
    v2f a = *(const v2f*)(arow + k);            // global_load_b64
    v2f b;
    b.x = bcol[(k + 0) * 16];                   // ds_load_b32
    b.y = bcol[(k + 1) * 16];
    acc = __builtin_amdgcn_wmma_f32_16x16x4_f32(
        /*neg_a=*/false, a, /*neg_b=*/false, b,
        /*c_mod=*/(short)0, acc, /*reuse_a=*/false, /*reuse_b=*/false);
  }

  const float bv = bias[n0 + ln];
  float* __restrict__ yp = Y + (size_t)(tile_m * 16 + 8 * hi) * N + n0 + ln;
#pragma unroll
  for (int r = 0; r < 8; ++r)
    yp[(size_t)r * N] = acc[r] + bv;
}

// ---------------------------------------------------------------------------
// In-place softmax over the last 16 elements (SUMP) per (b,q,h) row.
// ---------------------------------------------------------------------------
__global__ __launch_bounds__(256) void softmax16_kernel(float* __restrict__ attn,
                                                        int rows)
{
  const int r = blockIdx.x * 256 + threadIdx.x;
  if (r >= rows) return;
  float4* p = (float4*)(attn + (size_t)r * kSumP);
  float4 v0 = p[0], v1 = p[1], v2 = p[2], v3 = p[3];
  float v[16] = {v0.x, v0.y, v0.z, v0.w, v1.x, v1.y, v1.z, v1.w,
                 v2.x, v2.y, v2.z, v2.w, v3.x, v3.y, v3.z, v3.w};
  float m = v[0];
#pragma unroll
  for (int i = 1; i < 16; ++i) m = fmaxf(m, v[i]);
  float s = 0.0f;
#pragma unroll
  for (int i = 0; i < 16; ++i) { v[i] = __expf(v[i] - m); s += v[i]; }
  const float inv = 1.0f / s;
  p[0] = make_float4(v[0] * inv, v[1] * inv, v[2] * inv, v[3] * inv);
  p[1] = make_float4(v[4] * inv, v[5] * inv, v[6] * inv, v[7] * inv);
  p[2] = make_float4(v[8] * inv, v[9] * inv, v[10] * inv, v[11] * inv);
  p[3] = make_float4(v[12] * inv, v[13] * inv, v[14] * inv, v[15] * inv);
}

// ---------------------------------------------------------------------------
// Deformable sampling. One wave per (b,q,h); lane = head-dim channel (HD=32).
// Offsets / attention weights / reference points are wave-uniform broadcast
// loads; each bilinear corner fetch is one coalesced 128B transaction
// (v_proj layout [b, lv, 256], channel = h*32 + lane).
// out layout: [b, q, NH*HD] (= [b,q,256]).
// ---------------------------------------------------------------------------
__global__ __launch_bounds__(256) void msda_sample_kernel(
    const float* __restrict__ off,   // [BS,LQ,NH,SUMP,2]
    const float* __restrict__ attn,  // [BS,LQ,NH,SUMP] (post-softmax)
    const float* __restrict__ refp,  // [BS,LQ,1,4]
    const float* __restrict__ v,     // [BS,LV,256]
    float* __restrict__ out)         // [BS,LQ,256]
{
  const int gid  = blockIdx.x * 8 + (threadIdx.x >> 5);  // (b,q,h) index
  const int lane = threadIdx.x & 31;
  const int h = gid & (kNH - 1);
  const int q = (gid >> 3) & (kLQ - 1);
  const int b = gid >> 15;                               // / (NH*LQ)

  const size_t bq = (size_t)b * kLQ + q;
  const float4 rp = *(const float4*)(refp + bq * 4);
  const float* __restrict__ offp = off + (bq * kNH + h) * (kSumP * 2);
  const float* __restrict__ atp  = attn + (bq * kNH + h) * kSumP;
  const int ch = h * kHD + lane;

  constexpr int Hs[kLevels]     = {80, 40, 20, 10};
  constexpr int vstart[kLevels] = {0, 6400, 8000, 8400};

  float acc = 0.0f;
#pragma unroll
  for (int l = 0; l < kLevels; ++l) {
    const int Hl = Hs[l], Wl = Hs[l];                    // square levels
    const float* __restrict__ vb =
        v + ((size_t)b * kLV + vstart[l]) * kEmbed + ch;
#pragma unroll
    for (int p = 0; p < kP; ++p) {
      const int pg = l * kP + p;
      const float ox = offp[2 * pg];
      const float oy = offp[2 * pg + 1];
      const float w  = atp[pg];
      // loc = ref_xy + off * (1/P) * ref_wh * OFFSET_SCALE ; 1/P*0.5 = 0.125
      const float x = (rp.x + ox * 0.125f * rp.z) * (float)Wl - 0.5f;
      const float y = (rp.y + oy * 0.125f * rp.w) * (float)Hl - 0.5f;
      const float fx = floorf(x), fy = floorf(y);
      const int x0 = (int)fx, y0 = (int)fy;
      const float lx = x - fx, ly = y - fy;

      float s = 0.0f;
      auto corner = [&](int xi, int yi, float cw) {
        const bool valid = ((unsigned)xi < (unsigned)Wl) &
                           ((unsigned)yi < (unsigned)Hl);
        const int xc = iclamp(xi, 0, Wl - 1);
        const int yc = iclamp(yi, 0, Hl - 1);
        const float val = vb[(size_t)(yc * Wl + xc) * kEmbed];
        s = fmaf(valid ? cw : 0.0f, val, s);
      };
      corner(x0,     y0,     (1.0f - lx) * (1.0f - ly));
      corner(x0 + 1, y0,     lx * (1.0f - ly));
      corner(x0,     y0 + 1, (1.0f - lx) * ly);
      corner(x0 + 1, y0 + 1, lx * ly);
      acc = fmaf(w, s, acc);
    }
  }
  out[bq * kEmbed + ch] = acc;
}

// ---------------------------------------------------------------------------
extern "C" void kernel_launch(void* const* d_in, const int* in_sizes, int n_in,
                              void* d_out, int out_size, void* d_ws, size_t ws_size,
                              hipStream_t stream)
{
  (void)in_sizes; (void)n_in; (void)out_size; (void)ws_size;

  const float* query  = (const float*)d_in[0];
  const float* refp   = (const float*)d_in[1];
  const float* value  = (const float*)d_in[2];
  const float* W_off  = (const float*)d_in[3];
  const float* b_off  = (const float*)d_in[4];
  const float* W_attn = (const float*)d_in[5];
  const float* b_attn = (const float*)d_in[6];
  const float* W_v    = (const float*)d_in[7];
  const float* b_v    = (const float*)d_in[8];
  const float* W_o    = (const float*)d_in[9];
  const float* b_o    = (const float*)d_in[10];
  float* out = (float*)d_out;

  // workspace carve-up (~77 MB total)
  float* v_proj  = (float*)d_ws;                                   // BS*LV*256
  float* offb    = v_proj + (size_t)kBS * kLV * kEmbed;            // BS*LQ*256
  float* attnb   = offb + (size_t)kBS * kLQ * kEmbed;              // BS*LQ*128
  float* sampled = attnb + (size_t)kBS * kLQ * (kNH * kSumP);      // BS*LQ*256

  // 1) v = value @ W_v + b_v
  {
    const int M = kBS * kLV, N = kEmbed;                 // 34000 x 256
    dim3 grid(N / 16, (M / 16 + 7) / 8);
    gemm_k256_wmma<<<grid, 256, 0, stream>>>(value, W_v, b_v, v_proj, M, N);
  }
  // 2) off = query @ W_off + b_off
  {
    const int M = kBS * kLQ, N = kEmbed;                 // 16384 x 256
    dim3 grid(N / 16, (M / 16 + 7) / 8);
    gemm_k256_wmma<<<grid, 256, 0, stream>>>(query, W_off, b_off, offb, M, N);
  }
  // 3) attn logits = query @ W_attn + b_attn
  {
    const int M = kBS * kLQ, N = kNH * kSumP;            // 16384 x 128
    dim3 grid(N / 16, (M / 16 + 7) / 8);
    gemm_k256_wmma<<<grid, 256, 0, stream>>>(query, W_attn, b_attn, attnb, M, N);
  }
  // 4) softmax over SUMP=16
  {
    const int rows = kBS * kLQ * kNH;                    // 131072
    softmax16_kernel<<<(rows + 255) / 256, 256, 0, stream>>>(attnb, rows);
  }
  // 5) bilinear sampling + weighted accumulation
  {
    const int waves = kBS * kLQ * kNH;                   // 131072 waves
    msda_sample_kernel<<<waves / 8, 256, 0, stream>>>(offb, attnb, refp,
                                                      v_proj, sampled);
  }
  // 6) out = sampled @ W_o + b_o
  {
    const int M = kBS * kLQ, N = kEmbed;
    dim3 grid(N / 16, (M / 16 + 7) / 8);
    gemm_k256_wmma<<<grid, 256, 0, stream>>>(sampled, W_o, b_o, out, M, N);
  }
}